// Net_52432960750122
// MI455X (gfx1250) — compile-verified
//
#include <hip/hip_runtime.h>

typedef __attribute__((ext_vector_type(2))) float v2f;
typedef __attribute__((ext_vector_type(8))) float v8f;

#define GCN_NODES   100000
#define GCN_EDGES   1600000
#define GCN_IN      128
#define GCN_HID     128
#define GCN_OUT     64

// ---------------------------------------------------------------------------
// Degree accumulation: one thread per edge, f32 atomics (L2-resident counters)
// ---------------------------------------------------------------------------
__global__ __launch_bounds__(256) void gcn_deg_kernel(
    const int* __restrict__ src, const int* __restrict__ dst,
    float* __restrict__ out_deg, float* __restrict__ in_deg, int n_edges) {
  int e = blockIdx.x * blockDim.x + threadIdx.x;
  if (e < n_edges) {
    atomicAdd(out_deg + src[e], 1.0f);
    atomicAdd(in_deg + dst[e], 1.0f);
  }
}

// deg -> rsqrt(clamp(deg,1)) in place for both buffers
__global__ __launch_bounds__(256) void gcn_norm_kernel(
    float* __restrict__ out_n, float* __restrict__ in_n, int n_nodes) {
  int i = blockIdx.x * blockDim.x + threadIdx.x;
  if (i < n_nodes) {
    out_n[i] = rsqrtf(fmaxf(out_n[i], 1.0f));
    in_n[i]  = rsqrtf(fmaxf(in_n[i], 1.0f));
  }
}

// ---------------------------------------------------------------------------
// WMMA fp32 GEMM: H[M x N] = f(A)[M x K] @ W[K x N]
//   FUSE=false: f(A) = A * out_norm[row]                    (layer 1)
//   FUSE=true : f(A) = relu(A*in_norm[row] + b1[k]) * out_norm[row]  (layer 2)
// One wave per 16x16 C tile; K-loop of 4 with v_wmma_f32_16x16x4_f32.
// M is a multiple of 16 and grid is exact, so EXEC is all-ones in WMMA waves.
// ---------------------------------------------------------------------------
template <int KDIM, int NDIM, bool FUSE>
__global__ __launch_bounds__(256) void gcn_gemm_wmma(
    const float* __restrict__ A, const float* __restrict__ W,
    const float* __restrict__ out_norm, const float* __restrict__ in_norm,
    const float* __restrict__ bias, float* __restrict__ H, int M) {
  const int lane  = threadIdx.x & 31;
  const int wave  = (blockIdx.x * blockDim.x + threadIdx.x) >> 5;
  const int nTil  = NDIM / 16;
  const int mTile = wave / nTil;
  const int nTile = wave % nTil;
  if (mTile * 16 >= M) return;   // uniform per wave; never taken with exact grid

  const int row  = lane & 15;          // A row / B,C column within tile
  const int kgrp = (lane >> 4) << 1;   // lane-half K offset: 0 or 2
  const int m0   = mTile * 16;
  const int n0   = nTile * 16;
  const int gRow = m0 + row;

  const float onrm = out_norm[gRow];
  float inrm = 0.0f;
  if (FUSE) inrm = in_norm[gRow];

  const float* __restrict__ arow = A + (size_t)gRow * KDIM;
  const float* __restrict__ wcol = W + n0 + row;

  v8f c = {};
#pragma unroll
  for (int k = 0; k < KDIM; k += 4) {
    const int ka = k + kgrp;
    // A fragment (2 consecutive K elements per lane), norm/relu fused in
    float a0 = arow[ka];
    float a1 = arow[ka + 1];
    if (FUSE) {
      a0 = fmaxf(fmaf(a0, inrm, bias[ka]),     0.0f) * onrm;
      a1 = fmaxf(fmaf(a1, inrm, bias[ka + 1]), 0.0f) * onrm;
    } else {
      a0 *= onrm;
      a1 *= onrm;
    }
    v2f av; av[0] = a0; av[1] = a1;
    // B fragment: VGPR j holds W[(ka+j)*NDIM + n0 + (lane&15)]
    v2f bv;
    bv[0] = wcol[(size_t)ka * NDIM];
    bv[1] = wcol[(size_t)(ka + 1) * NDIM];
    c = __builtin_amdgcn_wmma_f32_16x16x4_f32(false, av, false, bv,
                                              (short)0, c, false, false);
  }

  // C layout: VGPR i -> row (i + (lane>>4)*8), col (lane&15)
  const int mOff = (lane >> 4) << 3;
  float* __restrict__ hout = H + (size_t)(m0 + mOff) * NDIM + (n0 + row);
#pragma unroll
  for (int i = 0; i < 8; ++i) hout[(size_t)i * NDIM] = c[i];
}

// ---------------------------------------------------------------------------
// Edge scatter: agg[dst[e]] += h[src[e]].  One wave per edge; each lane moves
// D/32 contiguous floats (128-bit load + f32 atomics, all hitting L2).
// ---------------------------------------------------------------------------
template <int D>
__global__ __launch_bounds__(256) void gcn_scatter_kernel(
    const float* __restrict__ h, const int* __restrict__ src,
    const int* __restrict__ dst, float* __restrict__ agg, int n_edges) {
  const int lane = threadIdx.x & 31;
  const int e    = (blockIdx.x * blockDim.x + threadIdx.x) >> 5;
  if (e >= n_edges) return;
  const int s = src[e];
  const int d = dst[e];
  const int base = lane * (D / 32);
  const float* __restrict__ hp = h   + (size_t)s * D + base;
  float* __restrict__       ap = agg + (size_t)d * D + base;
  if (D / 32 == 4) {
    float4 v = *(const float4*)hp;
    atomicAdd(ap + 0, v.x);
    atomicAdd(ap + 1, v.y);
    atomicAdd(ap + 2, v.z);
    atomicAdd(ap + 3, v.w);
  } else {
    float2 v = *(const float2*)hp;
    atomicAdd(ap + 0, v.x);
    atomicAdd(ap + 1, v.y);
  }
}

// out = out * in_norm[node] + b2[dim]   (in place over d_out)
__global__ __launch_bounds__(256) void gcn_finalize_kernel(
    float* __restrict__ out, const float* __restrict__ in_norm,
    const float* __restrict__ b2, int n_elems) {
  int i = blockIdx.x * blockDim.x + threadIdx.x;
  if (i < n_elems) {
    int node = i / GCN_OUT;
    int dim  = i % GCN_OUT;
    out[i] = fmaf(out[i], in_norm[node], b2[dim]);
  }
}

static inline size_t gcn_align(size_t x) { return (x + 255) & ~(size_t)255; }

extern "C" void kernel_launch(void* const* d_in, const int* in_sizes, int n_in,
                              void* d_out, int out_size, void* d_ws, size_t ws_size,
                              hipStream_t stream) {
  const float* features = (const float*)d_in[0];  // [100000 x 128]
  const int*   src      = (const int*)d_in[1];    // [1.6M]
  const int*   dst      = (const int*)d_in[2];    // [1.6M]
  const float* W1       = (const float*)d_in[3];  // [128 x 128]
  const float* b1       = (const float*)d_in[4];  // [128]
  const float* W2       = (const float*)d_in[5];  // [128 x 64]
  const float* b2       = (const float*)d_in[6];  // [64]
  float*       out      = (float*)d_out;          // [100000 x 64]

  const int nNodes = GCN_NODES;
  const int nEdges = GCN_EDGES;

  // Workspace carve-up
  const size_t sz_norm = gcn_align((size_t)nNodes * sizeof(float));           // 400 KB
  const size_t sz_h    = gcn_align((size_t)nNodes * GCN_HID * sizeof(float)); // 51.2 MB
  const size_t needed  = 2 * sz_norm + 2 * sz_h;                              // ~103 MB
  if (ws_size < needed) return;

  char* ws = (char*)d_ws;
  float* out_norm = (float*)(ws);
  float* in_norm  = (float*)(ws + sz_norm);
  float* h        = (float*)(ws + 2 * sz_norm);          // layer-1 projected msgs
  float* agg1     = (float*)(ws + 2 * sz_norm + sz_h);   // layer-1 aggregate

  // Zero accumulators (graph-capture-safe async memsets)
  hipMemsetAsync(out_norm, 0, (size_t)nNodes * sizeof(float), stream);
  hipMemsetAsync(in_norm,  0, (size_t)nNodes * sizeof(float), stream);
  hipMemsetAsync(agg1,     0, (size_t)nNodes * GCN_HID * sizeof(float), stream);
  hipMemsetAsync(out,      0, (size_t)nNodes * GCN_OUT * sizeof(float), stream);

  // 1) Degrees -> rsqrt norms
  gcn_deg_kernel<<<(nEdges + 255) / 256, 256, 0, stream>>>(src, dst, out_norm,
                                                           in_norm, nEdges);
  gcn_norm_kernel<<<(nNodes + 255) / 256, 256, 0, stream>>>(out_norm, in_norm,
                                                            nNodes);

  // 2) h = (features * out_norm) @ W1   — WMMA, 6250 M-tiles x 8 N-tiles
  {
    const int waves  = (nNodes / 16) * (GCN_HID / 16);  // 50000
    const int blocks = waves / 8;                       // 6250, exact
    gcn_gemm_wmma<GCN_IN, GCN_HID, false><<<blocks, 256, 0, stream>>>(
        features, W1, out_norm, nullptr, nullptr, h, nNodes);
  }

  // 3) agg1[dst] += h[src]   (wave per edge, 4 floats/lane)
  {
    const long long threads = (long long)nEdges * 32;
    gcn_scatter_kernel<GCN_HID><<<(int)((threads + 255) / 256), 256, 0, stream>>>(
        h, src, dst, agg1, nEdges);
  }

  // 4) h2 = (relu(agg1*in_norm + b1) * out_norm) @ W2 — fused activation GEMM
  float* h2 = h;  // reuse layer-1 message buffer
  {
    const int waves  = (nNodes / 16) * (GCN_OUT / 16);  // 25000
    const int blocks = waves / 8;                       // 3125, exact
    gcn_gemm_wmma<GCN_HID, GCN_OUT, true><<<blocks, 256, 0, stream>>>(
        agg1, W2, out_norm, in_norm, b1, h2, nNodes);
  }

  // 5) out[dst] += h2[src]   (accumulate straight into d_out, 2 floats/lane)
  {
    const long long threads = (long long)nEdges * 32;
    gcn_scatter_kernel<GCN_OUT><<<(int)((threads + 255) / 256), 256, 0, stream>>>(
        h2, src, dst, out, nEdges);
  }

  // 6) out = out * in_norm + b2
  {
    const int n = nNodes * GCN_OUT;
    gcn_finalize_kernel<<<(n + 255) / 256, 256, 0, stream>>>(out, in_norm, b2, n);
  }
}